// MambaBlock_57071525430014
// MI455X (gfx1250) — compile-verified
//
#include <hip/hip_runtime.h>
#include <hip/hip_bf16.h>

typedef __attribute__((ext_vector_type(2))) float v2f;
typedef __attribute__((ext_vector_type(8))) float v8f;

#define D_MODEL 1024
#define D_STATE 16
#define D_CONV  4
#define D_INNER 2048
#define DT_RANK 64
#define BB 2
#define LL 2048
#define TOKENS (BB * LL)   // 4096
#define XDBL_W (DT_RANK + 2 * D_STATE)  // 96

__device__ __forceinline__ float sigmoidf_(float x) { return 1.f / (1.f + __expf(-x)); }

// ---------------------------------------------------------------------------
// LayerNorm over D_MODEL=1024, one block (256 threads) per token.
// ---------------------------------------------------------------------------
__global__ __launch_bounds__(256)
void ln_kernel(const float* __restrict__ x, const float* __restrict__ w,
               const float* __restrict__ b, float* __restrict__ out)
{
    const int row = blockIdx.x;
    const float* xr = x + (size_t)row * D_MODEL;
    float v[4];
    float s = 0.f, s2 = 0.f;
#pragma unroll
    for (int i = 0; i < 4; ++i) {
        v[i] = xr[threadIdx.x + i * 256];
        s += v[i];
        s2 += v[i] * v[i];
    }
#pragma unroll
    for (int off = 16; off > 0; off >>= 1) {
        s  += __shfl_down(s,  off, 32);
        s2 += __shfl_down(s2, off, 32);
    }
    __shared__ float rs[8], rq[8];
    __shared__ float mu_s, rstd_s;
    const int wave = threadIdx.x >> 5;
    const int lane = threadIdx.x & 31;
    if (lane == 0) { rs[wave] = s; rq[wave] = s2; }
    __syncthreads();
    if (threadIdx.x == 0) {
        float a = 0.f, q = 0.f;
#pragma unroll
        for (int i = 0; i < 8; ++i) { a += rs[i]; q += rq[i]; }
        const float mu  = a * (1.f / D_MODEL);
        const float var = q * (1.f / D_MODEL) - mu * mu;
        mu_s = mu;
        rstd_s = rsqrtf(var + 1e-5f);
    }
    __syncthreads();
    const float mu = mu_s, rstd = rstd_s;
    float* orow = out + (size_t)row * D_MODEL;
#pragma unroll
    for (int i = 0; i < 4; ++i) {
        const int c = threadIdx.x + i * 256;
        orow[c] = (v[i] - mu) * rstd * w[c] + b[c];
    }
}

// ---------------------------------------------------------------------------
// fp32 GEMM via V_WMMA_F32_16X16X4_F32, software double-buffered.
//
// Wave tile: (16*MT) M x (16*NT) N -> MT*NT v8f accumulators.
// Block: 4 waves arranged 2 (M) x 2 (N) -> block tile (32*MT) x (32*NT).
// Row-major A [M,K] (lda), row-major B [K,N] (ldb), row-major C [M,N] (ldc).
// Optional epilogue: C = scale * (A@B) + residual.
//
// f32 16x16x4 VGPR layouts (ISA 7.12.2):
//   A: lane = (M%16) + 16*(K/2), vgpr v = K%2      -> per-lane v2f from A row
//   B: lane = (N%16) + 16*(K/2), vgpr v = K%2      (mirror of A)
//   C/D: lane = (N%16) + 16*(M/8), vgpr v = M%8
//
// Double-buffering: loads for step k+4 are issued before the MT*NT WMMAs of
// step k, so the s_wait_loadcnt lands after the MMA burst (on the register
// rotation), not in front of each WMMA.
// ---------------------------------------------------------------------------
template <int MT, int NT>
__global__ __launch_bounds__(128)
void wmma_gemm_kernel(const float* __restrict__ A, int lda,
                      const float* __restrict__ Bm, int ldb,
                      float* __restrict__ C, int ldc, int K,
                      const float* __restrict__ residual, int ldr,
                      const float* __restrict__ scale_ptr)
{
    const int lane = threadIdx.x & 31;
    const int wave = threadIdx.x >> 5;
    const int wy = wave & 1;            // M placement within block
    const int wx = wave >> 1;           // N placement within block
    const int m0 = blockIdx.y * (32 * MT) + wy * (16 * MT);
    const int n0 = blockIdx.x * (32 * NT) + wx * (16 * NT);
    const int lm = lane & 15;           // M (for A) / N (for B) within tile
    const int kh = (lane >> 4) << 1;    // K half-select: 0 or 2

    v8f acc[MT][NT];
#pragma unroll
    for (int i = 0; i < MT; ++i)
#pragma unroll
        for (int t = 0; t < NT; ++t)
#pragma unroll
            for (int e = 0; e < 8; ++e) acc[i][t][e] = 0.f;

    const float* Ap = A + (size_t)(m0 + lm) * lda + kh;
    const float* Bp = Bm + (size_t)kh * ldb + n0 + lm;

    auto loadA = [&](int k, v2f (&a)[MT]) {
#pragma unroll
        for (int i = 0; i < MT; ++i)
            a[i] = *(const v2f*)(Ap + (size_t)i * 16 * lda + k);
    };
    auto loadB = [&](int k, v2f (&b)[NT]) {
        const float* r0 = Bp + (size_t)k * ldb;
        const float* r1 = r0 + ldb;
#pragma unroll
        for (int t = 0; t < NT; ++t) {
            b[t].x = r0[t * 16];
            b[t].y = r1[t * 16];
        }
    };

    v2f a[MT], b[NT], a2[MT], b2[NT];
    loadA(0, a);
    loadB(0, b);

    int k = 0;
    for (; k + 4 < K; k += 4) {
        // issue next-step loads first (consumed only after this step's MMAs)
        loadA(k + 4, a2);
        loadB(k + 4, b2);
        // stream-ahead prefetch into all cache levels (speculative, OOB-safe)
        __builtin_prefetch(Ap + k + 128, 0, 3);
        __builtin_prefetch(Bp + (size_t)(k + 16) * ldb, 0, 3);
#pragma unroll
        for (int i = 0; i < MT; ++i)
#pragma unroll
            for (int t = 0; t < NT; ++t)
                acc[i][t] = __builtin_amdgcn_wmma_f32_16x16x4_f32(
                    false, a[i], false, b[t], (short)0, acc[i][t], false, false);
#pragma unroll
        for (int i = 0; i < MT; ++i) a[i] = a2[i];
#pragma unroll
        for (int t = 0; t < NT; ++t) b[t] = b2[t];
    }
    // tail step
#pragma unroll
    for (int i = 0; i < MT; ++i)
#pragma unroll
        for (int t = 0; t < NT; ++t)
            acc[i][t] = __builtin_amdgcn_wmma_f32_16x16x4_f32(
                false, a[i], false, b[t], (short)0, acc[i][t], false, false);

    const float scale = scale_ptr ? scale_ptr[0] : 1.0f;
    const int mb = (lane >> 4) << 3;
#pragma unroll
    for (int i = 0; i < MT; ++i) {
#pragma unroll
        for (int t = 0; t < NT; ++t) {
            const int n = n0 + t * 16 + lm;
#pragma unroll
            for (int v = 0; v < 8; ++v) {
                const int m = m0 + i * 16 + mb + v;
                float val = acc[i][t][v] * scale;
                if (residual) val += residual[(size_t)m * ldr + n];
                C[(size_t)m * ldc + n] = val;
            }
        }
    }
}

// ---------------------------------------------------------------------------
// Causal depthwise conv1d (K=4) + bias + SiLU.
// x_main lives in xz buffer: row stride 2*D_INNER, columns [0, D_INNER).
// ---------------------------------------------------------------------------
__global__ __launch_bounds__(256)
void conv_silu_kernel(const float* __restrict__ xz, const float* __restrict__ w,
                      const float* __restrict__ bias, float* __restrict__ out)
{
    const size_t idx = (size_t)blockIdx.x * 256 + threadIdx.x;  // over B*L*Di
    const int c = (int)(idx & (D_INNER - 1));
    const size_t bl = idx >> 11;           // /D_INNER
    const int l = (int)(bl & (LL - 1));
    const int b = (int)(bl >> 11);
    float acc = bias[c];
#pragma unroll
    for (int k = 0; k < D_CONV; ++k) {
        const int ls = l - (D_CONV - 1) + k;
        if (ls >= 0)
            acc += w[c * D_CONV + k] *
                   xz[((size_t)(b * LL + ls)) * (2 * D_INNER) + c];
    }
    out[idx] = acc * sigmoidf_(acc);
}

// new_conv_state[b, c, j] = x_main[b, L-3+j, c]
__global__ __launch_bounds__(256)
void conv_state_kernel(const float* __restrict__ xz, float* __restrict__ out)
{
    const int idx = blockIdx.x * 256 + threadIdx.x;
    if (idx >= BB * D_INNER * (D_CONV - 1)) return;
    const int j = idx % (D_CONV - 1);
    const int c = (idx / (D_CONV - 1)) % D_INNER;
    const int b = idx / ((D_CONV - 1) * D_INNER);
    out[idx] = xz[((size_t)(b * LL + (LL - (D_CONV - 1) + j))) * (2 * D_INNER) + c];
}

// dt = softplus(dt_lin + bias)  (in place)
__global__ __launch_bounds__(256)
void dt_softplus_kernel(float* __restrict__ dt, const float* __restrict__ bias)
{
    const size_t idx = (size_t)blockIdx.x * 256 + threadIdx.x;
    const int d = (int)(idx & (D_INNER - 1));
    const float v = dt[idx] + bias[d];
    dt[idx] = (v > 20.f) ? v : __logf(1.f + __expf(v));
}

// ---------------------------------------------------------------------------
// Selective scan: one lane per (b, d) channel, h[16] in registers,
// sequential over L. B/C rows broadcast from L0 via float4 loads.
// ---------------------------------------------------------------------------
__global__ __launch_bounds__(256)
void scan_kernel(const float* __restrict__ dt, const float* __restrict__ xc,
                 const float* __restrict__ xdbl, const float* __restrict__ A_log,
                 float* __restrict__ y, float* __restrict__ ssm_out)
{
    const int t = blockIdx.x * 256 + threadIdx.x;  // 0..B*Di-1
    const int d = t & (D_INNER - 1);
    const int b = t >> 11;

    float Aval[D_STATE];
#pragma unroll
    for (int n = 0; n < D_STATE; ++n)
        Aval[n] = -__expf(A_log[d * D_STATE + n]);

    float h[D_STATE];
#pragma unroll
    for (int n = 0; n < D_STATE; ++n) h[n] = 0.f;

    for (int l = 0; l < LL; ++l) {
        const size_t roff = (size_t)(b * LL + l);
        const float dtv = dt[roff * D_INNER + d];
        const float xv  = xc[roff * D_INNER + d];
        const float4* Bp = (const float4*)(xdbl + roff * XDBL_W + DT_RANK);
        const float4* Cp = (const float4*)(xdbl + roff * XDBL_W + DT_RANK + D_STATE);
        float Bb[D_STATE], Cc[D_STATE];
#pragma unroll
        for (int q = 0; q < 4; ++q) {
            const float4 bq = Bp[q];
            const float4 cq = Cp[q];
            Bb[4*q+0] = bq.x; Bb[4*q+1] = bq.y; Bb[4*q+2] = bq.z; Bb[4*q+3] = bq.w;
            Cc[4*q+0] = cq.x; Cc[4*q+1] = cq.y; Cc[4*q+2] = cq.z; Cc[4*q+3] = cq.w;
        }
        const float bx = dtv * xv;
        float yv = 0.f;
#pragma unroll
        for (int n = 0; n < D_STATE; ++n) {
            const float dA = __expf(Aval[n] * dtv);
            h[n] = dA * h[n] + bx * Bb[n];
            yv += h[n] * Cc[n];
        }
        y[roff * D_INNER + d] = yv;
    }

#pragma unroll
    for (int n = 0; n < D_STATE; ++n)
        ssm_out[(size_t)(b * D_INNER + d) * D_STATE + n] = h[n];
}

// y = (y + xc * D) * silu(z), in place in y
__global__ __launch_bounds__(256)
void gate_kernel(float* __restrict__ y, const float* __restrict__ xc,
                 const float* __restrict__ xz, const float* __restrict__ Dp)
{
    const size_t idx = (size_t)blockIdx.x * 256 + threadIdx.x;
    const int d = (int)(idx & (D_INNER - 1));
    const size_t row = idx >> 11;
    const float z  = xz[row * (2 * D_INNER) + D_INNER + d];
    const float yv = y[idx] + xc[idx] * Dp[d];
    y[idx] = yv * z * sigmoidf_(z);
}

// ---------------------------------------------------------------------------
extern "C" void kernel_launch(void* const* d_in, const int* in_sizes, int n_in,
                              void* d_out, int out_size, void* d_ws, size_t ws_size,
                              hipStream_t stream)
{
    const float* x         = (const float*)d_in[0];
    const float* ln_w      = (const float*)d_in[1];
    const float* ln_b      = (const float*)d_in[2];
    const float* in_proj_w = (const float*)d_in[3];
    const float* conv_w    = (const float*)d_in[4];
    const float* conv_b    = (const float*)d_in[5];
    const float* x_proj_w  = (const float*)d_in[6];
    const float* dt_proj_w = (const float*)d_in[7];
    const float* dt_proj_b = (const float*)d_in[8];
    const float* A_log     = (const float*)d_in[9];
    const float* D_param   = (const float*)d_in[10];
    const float* out_projw = (const float*)d_in[11];
    const float* res_scale = (const float*)d_in[12];

    float* out      = (float*)d_out;                       // [B,L,D]   4,194,304
    float* out_ssm  = out + (size_t)BB * LL * D_MODEL;     // [B,Di,N]     65,536
    float* out_conv = out_ssm + (size_t)BB * D_INNER * D_STATE; // [B,Di,3] 12,288

    float* ws    = (float*)d_ws;
    float* xn    = ws;                                   //  4,194,304 f
    float* xz    = xn    + (size_t)TOKENS * D_MODEL;     // 16,777,216 f
    float* xc    = xz    + (size_t)TOKENS * 2 * D_INNER; //  8,388,608 f
    float* xdbl  = xc    + (size_t)TOKENS * D_INNER;     //    393,216 f
    float* dtbuf = xdbl  + (size_t)TOKENS * XDBL_W;      //  8,388,608 f
    float* ybuf  = dtbuf + (size_t)TOKENS * D_INNER;     //  8,388,608 f

    // 1. LayerNorm
    ln_kernel<<<TOKENS, 256, 0, stream>>>(x, ln_w, ln_b, xn);

    // 2. in_proj: [4096,1024] @ [1024,4096] -> xz   (block tile 64x128)
    wmma_gemm_kernel<2, 4><<<dim3((2 * D_INNER) / 128, TOKENS / 64), 128, 0, stream>>>(
        xn, D_MODEL, in_proj_w, 2 * D_INNER, xz, 2 * D_INNER, D_MODEL,
        nullptr, 0, nullptr);

    // 3. causal depthwise conv + SiLU; conv tail state
    conv_silu_kernel<<<(TOKENS * D_INNER) / 256, 256, 0, stream>>>(
        xz, conv_w, conv_b, xc);
    conv_state_kernel<<<(BB * D_INNER * (D_CONV - 1) + 255) / 256, 256, 0, stream>>>(
        xz, out_conv);

    // 4. x_proj: [4096,2048] @ [2048,96] -> xdbl   (block tile 64x96)
    wmma_gemm_kernel<2, 3><<<dim3(XDBL_W / 96, TOKENS / 64), 128, 0, stream>>>(
        xc, D_INNER, x_proj_w, XDBL_W, xdbl, XDBL_W, D_INNER,
        nullptr, 0, nullptr);

    // 5. dt_proj: [4096,64] @ [64,2048] -> dtbuf  (A = xdbl cols 0..63, lda=96)
    wmma_gemm_kernel<2, 4><<<dim3(D_INNER / 128, TOKENS / 64), 128, 0, stream>>>(
        xdbl, XDBL_W, dt_proj_w, D_INNER, dtbuf, D_INNER, DT_RANK,
        nullptr, 0, nullptr);

    // 6. dt = softplus(dt + bias)
    dt_softplus_kernel<<<(TOKENS * D_INNER) / 256, 256, 0, stream>>>(dtbuf, dt_proj_b);

    // 7. selective scan (sequential over L, parallel over B*Di lanes)
    scan_kernel<<<(BB * D_INNER) / 256, 256, 0, stream>>>(
        dtbuf, xc, xdbl, A_log, ybuf, out_ssm);

    // 8. y = (y + xc*D) * silu(z)
    gate_kernel<<<(TOKENS * D_INNER) / 256, 256, 0, stream>>>(ybuf, xc, xz, D_param);

    // 9. out = res_scale * (y @ out_proj) + residual
    wmma_gemm_kernel<2, 4><<<dim3(D_MODEL / 128, TOKENS / 64), 128, 0, stream>>>(
        ybuf, D_INNER, out_projw, D_MODEL, out, D_MODEL, D_INNER,
        x, D_MODEL, res_scale);
}